// LineFinderLoss_35244501631343
// MI455X (gfx1250) — compile-verified
//
#include <hip/hip_runtime.h>
#include <hip/hip_bf16.h>

#define ALPHA 0.1f
#define EPSV  1e-5f
#define NB 384
#define NP 256     // predictions per batch (LSA columns)
#define NL 64      // max labels per batch (LSA rows)
#define CSTRIDE 260  // row stride (floats): %64==4 -> 16 rows hit distinct banks; 16B aligned

typedef __attribute__((ext_vector_type(2))) float v2f;
typedef __attribute__((ext_vector_type(8))) float v8f;

#if __has_builtin(__builtin_amdgcn_wmma_f32_16x16x4_f32)
#define HAS_WMMA_F32X4 1
#else
#define HAS_WMMA_F32X4 0
#endif

// LDS layout (floats then ints); all sections start 16B-aligned.
#define F_COSTT   (NL*CSTRIDE)   // 16640
#define F_APACK   (NP*4)         // 1024
#define F_BPACK   (NL*4)         // 256
#define SMEM_FLOATS (F_COSTT + F_APACK + F_BPACK + 3*NP /*cinit,logc,logca*/ \
                     + 3*NP /*px,py,pz*/ + NL /*u*/ + NP /*shortest*/)
#define SMEM_INTS   (NP /*prevA*/ + NP /*row4col*/ + NL /*col4row*/ + NP /*matched*/)
#define SMEM_BYTES  ((size_t)(SMEM_FLOATS + SMEM_INTS) * 4)

__global__ void linefinder_zero4(float* out) {
    if (threadIdx.x < 4) out[threadIdx.x] = 0.0f;
}

__global__ __launch_bounds__(32)
void linefinder_kernel(const float* __restrict__ pred,
                       const float* __restrict__ label,
                       const int* __restrict__ label_len,
                       float* __restrict__ out)
{
    extern __shared__ float smem[];
    float* costT    = smem;                      // [NL][CSTRIDE] cost^T: rows=labels, cols=preds
    float* apack    = costT + F_COSTT;           // [NP][4] = {-a*px, -a*py, -a*pz, a/2}
    float* bpack    = apack + F_APACK;           // [NL][4] = {lx, ly, lz, |l|^2}
    float* cinit    = bpack + F_BPACK;           // [NP] a/2*|p|^2 - log_c + log_ca
    float* logc     = cinit + NP;
    float* logca    = logc + NP;
    float* px       = logca + NP;
    float* py       = px + NP;
    float* pz       = py + NP;
    float* u        = pz + NP;                   // [NL] row duals
    float* shortest = u + NL;                    // [NP] (dumped once per augmentation)
    int*   prevA    = (int*)(shortest + NP);     // [NP]
    int*   row4col  = prevA + NP;                // [NP]
    int*   col4row  = row4col + NP;              // [NL]
    int*   matched  = col4row + NL;              // [NP]

    const int b = blockIdx.x;
    const int t = threadIdx.x;                   // 0..31, one wave per block
    const int L = label_len[b];

    // ---- Phase 0: pack per-pred / per-label terms into LDS ----
    const float4* pb4 = (const float4*)(pred + (size_t)b * NP * 4);
    #pragma unroll
    for (int k = 0; k < 8; ++k) {
        int p = t + 32 * k;
        float4 q = pb4[p];                       // global_load_b128
        px[p] = q.x; py[p] = q.y; pz[p] = q.z;
        ((float4*)apack)[p] = make_float4(-ALPHA*q.x, -ALPHA*q.y, -ALPHA*q.z, 0.5f*ALPHA);
        float lc  = logf(q.w + EPSV);
        float lca = logf(1.0f - q.w + EPSV);
        logc[p] = lc; logca[p] = lca;
        cinit[p] = 0.5f * ALPHA * (q.x*q.x + q.y*q.y + q.z*q.z) - lc + lca;
    }
    const float* lb = label + (size_t)b * NL * 3;
    #pragma unroll
    for (int k = 0; k < 2; ++k) {
        int l = 2*t + k;
        float x = lb[l*3+0], y = lb[l*3+1], z = lb[l*3+2];
        ((float4*)bpack)[l] = make_float4(x, y, z, x*x + y*y + z*z);
    }
    __syncthreads();

    // ---- Phase 1: cost matrix via V_WMMA_F32_16X16X4_F32 ----
    // D = A x B + C : A=alpha*[-p0,-p1,-p2,1/2] (16x4), B=[l0;l1;l2;|l|^2] (4x16),
    // C = alpha/2*|p|^2 - log_c + log_ca  ->  D = alpha/2*|p-l|^2 - log_c + log_ca
    // Reuse ordering: all 4 B tiles held in registers; A and C hoisted per pred-tile.
    const int hf = t >> 4;                       // half-wave: selects K-pair {0,1} or {2,3}
    const int lm = t & 15;                       // A: M index, B: N index
    const v2f* ap2 = (const v2f*)apack;          // granule index = p*2 + hf
    const v2f* bp2 = (const v2f*)bpack;
    v2f Bv[NL/16];
    #pragma unroll
    for (int lt = 0; lt < NL/16; ++lt)
        Bv[lt] = bp2[(lt*16 + lm)*2 + hf];       // 4x ds_load_b64, resident all phase
    for (int pt = 0; pt < NP/16; ++pt) {
        v2f Av = ap2[(pt*16 + lm)*2 + hf];       // 1x ds_load_b64 per pred-tile
        const float4* ci4 = (const float4*)(cinit + pt*16 + 8*hf);
        float4 c0 = ci4[0], c1 = ci4[1];         // 2x ds_load_b128 per pred-tile
        v8f Cv = {c0.x, c0.y, c0.z, c0.w, c1.x, c1.y, c1.z, c1.w};
        #pragma unroll
        for (int lt = 0; lt < NL/16; ++lt) {
#if HAS_WMMA_F32X4
            v8f Dv = __builtin_amdgcn_wmma_f32_16x16x4_f32(
                false, Av, false, Bv[lt], (short)0, Cv, false, false);
#else
            v8f Dv;  // scalar fallback (same math)
            #pragma unroll
            for (int r = 0; r < 8; ++r) {
                int pm = pt*16 + r + 8*hf, ln = lt*16 + lm;
                Dv[r] = Cv[r] + ALPHA*(-px[pm]*bpack[ln*4+0] - py[pm]*bpack[ln*4+1]
                                       - pz[pm]*bpack[ln*4+2]) + 0.5f*ALPHA*bpack[ln*4+3];
            }
#endif
            float4* dst = (float4*)(costT + (lt*16 + lm)*CSTRIDE + pt*16 + 8*hf);
            dst[0] = make_float4(Dv[0], Dv[1], Dv[2], Dv[3]);   // 2x ds_store_b128
            dst[1] = make_float4(Dv[4], Dv[5], Dv[6], Dv[7]);
        }
    }
    __syncthreads();

    // ---- Phase 2: Jonker-Volgenant SAP; per-lane register state for owned columns ----
    // Lane t owns columns j = t*8 + k (k=0..7) and rows l = 2t + k (k=0..1).
    float vv[8];                                 // column duals (registers)
    #pragma unroll
    for (int k = 0; k < 8; ++k) { vv[k] = 0.0f; row4col[t*8 + k] = -1; }
    #pragma unroll
    for (int k = 0; k < 2; ++k) { u[2*t + k] = 0.0f; col4row[2*t + k] = -1; }
    __syncthreads();

    const float INF = 3.4e38f;
    for (int cur = 0; cur < L; ++cur) {
        float sh[8]; int pv[8];
        unsigned scm = 0u, srm = 0u;             // SC (8 cols) / SR (2 rows) bitmasks
        #pragma unroll
        for (int k = 0; k < 8; ++k) { sh[k] = INF; pv[k] = -1; }

        float minVal = 0.0f;
        int i = cur, sink = -1;
        while (sink < 0) {
            if (t == (i >> 1)) srm |= 1u << (i & 1);      // SR[i]=1 (register)
            float base = minVal - u[i];                   // uniform ds_load
            const float4* cr4 = (const float4*)(costT + i*CSTRIDE + t*8);
            float4 ca = cr4[0], cb = cr4[1];              // two ds_load_b128
            float cr[8] = {ca.x, ca.y, ca.z, ca.w, cb.x, cb.y, cb.z, cb.w};
            float best = INF; int bestj = NP;
            #pragma unroll
            for (int k = 0; k < 8; ++k) {
                if (!(scm & (1u << k))) {
                    float d = base + cr[k] - vv[k];
                    if (d < sh[k]) { sh[k] = d; pv[k] = i; }
                    int j = t*8 + k;
                    if (sh[k] < best || (sh[k] == best && j < bestj)) { best = sh[k]; bestj = j; }
                }
            }
            // wave32 min+argmin reduce, tie -> lowest j (matches np.argmin)
            for (int off = 16; off > 0; off >>= 1) {
                float ob = __shfl_xor(best, off, 32);
                int   oj = __shfl_xor(bestj, off, 32);
                if (ob < best || (ob == best && oj < bestj)) { best = ob; bestj = oj; }
            }
            minVal = best;
            int jstar = bestj;
            if (t == (jstar >> 3)) scm |= 1u << (jstar & 7);   // SC[jstar]=1 (register)
            int rc = row4col[jstar];                           // uniform ds_load
            if (rc < 0) sink = jstar; else i = rc;
        }

        // dump owned shortest/prev to LDS (once per augmentation)
        *(float4*)(shortest + t*8)     = make_float4(sh[0], sh[1], sh[2], sh[3]);
        *(float4*)(shortest + t*8 + 4) = make_float4(sh[4], sh[5], sh[6], sh[7]);
        *(int4*)(prevA + t*8)     = make_int4(pv[0], pv[1], pv[2], pv[3]);
        *(int4*)(prevA + t*8 + 4) = make_int4(pv[4], pv[5], pv[6], pv[7]);
        __syncthreads();

        // dual updates (use col4row BEFORE augmentation, as in reference)
        if (t == 0) u[cur] += minVal;
        #pragma unroll
        for (int k = 0; k < 2; ++k) {
            int l = 2*t + k;
            if (l < L && l != cur && ((srm >> k) & 1u))
                u[l] += minVal - shortest[col4row[l]];
        }
        #pragma unroll
        for (int k = 0; k < 8; ++k)
            if (scm & (1u << k)) vv[k] -= minVal - sh[k];     // v-update in registers
        __syncthreads();

        // augment along the path (scalar, lane 0)
        if (t == 0) {
            int j = sink;
            while (true) {
                int ii = prevA[j];
                row4col[j] = ii;
                int tmp = col4row[ii];
                col4row[ii] = j;
                j = tmp;
                if (ii == cur) break;
            }
        }
        __syncthreads();
    }

    // ---- Phase 3: loss reduction ----
    #pragma unroll
    for (int k = 0; k < 8; ++k) matched[t*8 + k] = 0;
    __syncthreads();
    float loc = 0.0f, cpos = 0.0f;
    #pragma unroll
    for (int k = 0; k < 2; ++k) {
        int l = 2*t + k;
        if (l < L) {
            int p = col4row[l];
            matched[p] = 1;
            float dx = px[p] - bpack[l*4+0];
            float dy = py[p] - bpack[l*4+1];
            float dz = pz[p] - bpack[l*4+2];
            loc  += dx*dx + dy*dy + dz*dz;
            cpos -= logc[p];
        }
    }
    __syncthreads();
    float cneg = 0.0f;
    #pragma unroll
    for (int k = 0; k < 8; ++k) {
        int j = t*8 + k;
        if (!matched[j]) cneg -= logca[j];
    }
    for (int off = 16; off > 0; off >>= 1) {
        loc  += __shfl_xor(loc,  off, 32);
        cpos += __shfl_xor(cpos, off, 32);
        cneg += __shfl_xor(cneg, off, 32);
    }
    if (t == 0) {
        float aloc = ALPHA * 0.5f * loc;          // alpha * location_loss (this batch)
        atomicAdd(&out[1], aloc);
        atomicAdd(&out[0], aloc + cpos + cneg + 0.0001f);
        if (b == NB - 1) { out[2] = cpos; out[3] = cneg; }
    }
}

extern "C" void kernel_launch(void* const* d_in, const int* in_sizes, int n_in,
                              void* d_out, int out_size, void* d_ws, size_t ws_size,
                              hipStream_t stream) {
    (void)in_sizes; (void)n_in; (void)out_size; (void)d_ws; (void)ws_size;
    const float* pred      = (const float*)d_in[0];   // (B, 256, 4) f32
    const float* label     = (const float*)d_in[1];   // (B, 64, 3)  f32
    const int*   label_len = (const int*)d_in[2];     // (B,)        i32
    float* out = (float*)d_out;                       // 4 f32 scalars

    // ~82 KB dynamic LDS per block (gfx1250 WGP has 320 KB) — raise the cap.
    hipFuncSetAttribute((const void*)linefinder_kernel,
                        hipFuncAttributeMaxDynamicSharedMemorySize, (int)SMEM_BYTES);

    linefinder_zero4<<<1, 32, 0, stream>>>(out);
    linefinder_kernel<<<NB, 32, SMEM_BYTES, stream>>>(pred, label, label_len, out);
}